// CRF_19421842112851
// MI455X (gfx1250) — compile-verified
//
#include <hip/hip_runtime.h>
#include <hip/hip_bf16.h>

typedef float v2f __attribute__((ext_vector_type(2)));
typedef float v8f __attribute__((ext_vector_type(8)));

#define START_TAG 6
#define STOP_TAG  7

// ---------------------------------------------------------------------------
// Setup: build exp-space A fragments (WMMA 16x4 f32 A layout) into ws.
// ws layout (floats):
//   [  0.. 31] A1.v0   (K = 0 lanes<16 / K = 2 lanes>=16)
//   [ 32.. 63] A1.v1   (K = 1 / K = 3)
//   [ 64.. 95] A2.v0   (K = 4 / K = 6)
//   [ 96..127] A2.v1   (K = 5 / K = 7)
//   [128]      mmax (global max over both blended matrices)
//   [129..136] stop row: M_unch[STOP, j] (raw log-space)
//   [256..256+B) gold path scores
// A rows M=0..7 -> W'_unch output-state j=M ; rows M=8..15 -> W'_ch j=M-8.
// ---------------------------------------------------------------------------
__global__ __launch_bounds__(32) void crf_setup_kernel(
    const float* __restrict__ Ti, const float* __restrict__ Ta,
    const float* __restrict__ wui_, const float* __restrict__ wua_,
    const float* __restrict__ wci_, const float* __restrict__ wca_,
    float* __restrict__ wsA)
{
    const int lane = threadIdx.x;  // 0..31
    const float wui = wui_[0], wua = wua_[0], wci = wci_[0], wca = wca_[0];

    float mmax = -3.4e38f;
    for (int i = 0; i < 64; ++i) {
        float mu = wui * Ti[i] + wua * Ta[i];
        float mc = wci * Ti[i] + wca * Ta[i];
        mmax = fmaxf(mmax, fmaxf(mu, mc));
    }

    const int  row   = lane & 15;       // A-matrix row M (lane and lane+16 share M)
    const int  j     = row & 7;         // output state
    const bool chrow = row >= 8;        // rows 8..15 = "changed" matrix
    const int  kbase = (lane < 16) ? 0 : 2;

    for (int frag = 0; frag < 2; ++frag) {
        for (int vr = 0; vr < 2; ++vr) {
            int k = frag * 4 + kbase + vr;   // input state
            float m = chrow ? (wci * Ti[j * 8 + k] + wca * Ta[j * 8 + k])
                            : (wui * Ti[j * 8 + k] + wua * Ta[j * 8 + k]);
            wsA[frag * 64 + vr * 32 + lane] = __expf(m - mmax);
        }
    }
    if (lane == 0) wsA[128] = mmax;
    if (lane < 8)  wsA[129 + lane] = wui * Ti[STOP_TAG * 8 + lane] + wua * Ta[STOP_TAG * 8 + lane];
}

// ---------------------------------------------------------------------------
// Gold path score: one block per sequence b, LDS tree reduction.
// ---------------------------------------------------------------------------
__global__ __launch_bounds__(256) void crf_gold_kernel(
    const float* __restrict__ feats, const int* __restrict__ spk,
    const int* __restrict__ emos,
    const float* __restrict__ Ti, const float* __restrict__ Ta,
    const float* __restrict__ wui_, const float* __restrict__ wua_,
    const float* __restrict__ wci_, const float* __restrict__ wca_,
    float* __restrict__ gold, int T)
{
    __shared__ float red[256];
    const int b   = blockIdx.x;
    const int tid = threadIdx.x;
    const float wui = wui_[0], wua = wua_[0], wci = wci_[0], wca = wca_[0];

    const int base = b * T;
    float acc = 0.0f;
    for (int t = tid; t < T; t += 256) {
        int e  = emos[base + t];
        int ep = (t == 0) ? START_TAG : emos[base + t - 1];
        int ch = spk[base + t];
        float ti = Ti[e * 8 + ep], ta = Ta[e * 8 + ep];
        float tv = (ch > 0) ? (wci * ti + wca * ta) : (wui * ti + wua * ta);
        acc += tv + feats[((size_t)base + t) * 8 + e];
    }
    red[tid] = acc;
    __syncthreads();
    for (int s = 128; s > 0; s >>= 1) {
        if (tid < s) red[tid] += red[tid + s];
        __syncthreads();
    }
    if (tid == 0) {
        int el = emos[base + T - 1];
        gold[b] = red[0] + wui * Ti[STOP_TAG * 8 + el] + wua * Ta[STOP_TAG * 8 + el];
    }
}

// ---------------------------------------------------------------------------
// One CRF forward step in scaled-prob space (no normalization).
// ---------------------------------------------------------------------------
__device__ __forceinline__ void crf_step(const v2f a1, const v2f a2,
                                         v2f& b1, v2f& b2, int ch,
                                         float2 flo, float2 fhi,
                                         int m, bool hi)
{
    // D[j, n] = sum_k W'_{unch|ch}[j,k] * P[n,k]
    v8f d = {};
    d = __builtin_amdgcn_wmma_f32_16x16x4_f32(false, a1, false, b1,
                                              (short)0, d, false, false);
    d = __builtin_amdgcn_wmma_f32_16x16x4_f32(false, a2, false, b2,
                                              (short)0, d, false, false);

    // flag-select (unch result: lanes 0..15, ch result: lanes 16..31) and
    // transpose back into B-fragment form in one bpermute pass
    const int srcl = m + ((ch > 0) ? 16 : 0);
    float s0 = __shfl(d[0], srcl, 32), s2 = __shfl(d[2], srcl, 32);
    float s1 = __shfl(d[1], srcl, 32), s3 = __shfl(d[3], srcl, 32);
    float s4 = __shfl(d[4], srcl, 32), s6 = __shfl(d[6], srcl, 32);
    float s5 = __shfl(d[5], srcl, 32), s7 = __shfl(d[7], srcl, 32);

    // emission in prob space: this lane owns states {k0,k0+1,k0+4,k0+5}
    b1.x = (hi ? s2 : s0) * __expf(flo.x);
    b1.y = (hi ? s3 : s1) * __expf(flo.y);
    b2.x = (hi ? s6 : s4) * __expf(fhi.x);
    b2.y = (hi ? s7 : s5) * __expf(fhi.y);
}

// ---------------------------------------------------------------------------
// Forward algorithm: one wave per 16-row batch tile.
// State P kept as WMMA B fragment (B[k,n] = P[batch n, state k]):
//   b1.v0: k=0 (lanes<16) / k=2 (lanes>=16);  b1.v1: k=1 / k=3
//   b2.v0: k=4 / k=6;                         b2.v1: k=5 / k=7
// ---------------------------------------------------------------------------
__global__ __launch_bounds__(128) void crf_fwd_kernel(
    const float* __restrict__ feats, const int* __restrict__ spk,
    const float* __restrict__ wsA, const float* __restrict__ gold,
    float* __restrict__ out, int B, int T)
{
    const int  lane = threadIdx.x & 31;
    const int  wave = blockIdx.x * (blockDim.x >> 5) + (threadIdx.x >> 5);
    if (wave * 16 >= B) return;                 // wave-uniform

    const int  m   = lane & 15;                 // batch row within tile
    const bool hi  = lane >= 16;
    const int  k0  = hi ? 2 : 0;                // first state this lane owns
    int b = wave * 16 + m;
    const bool valid = (b < B);
    if (b >= B) b = B - 1;                      // clamp addresses only

    // constant exp-space transition fragments
    v2f a1, a2;
    a1.x = wsA[0 + lane];  a1.y = wsA[32 + lane];
    a2.x = wsA[64 + lane]; a2.y = wsA[96 + lane];
    const float mmax = wsA[128];

    // P0 = one-hot at START(=6): state 6 lives in b2.v0 of hi lanes
    v2f b1; b1.x = 0.0f; b1.y = 0.0f;
    v2f b2; b2.x = hi ? 1.0f : 0.0f; b2.y = 0.0f;
    float sumlog2 = 0.0f;

    const float* fbase = feats + (size_t)b * T * 8 + k0;   // lane's emission base
    const int*   sptr  = spk   + (size_t)b * T;

#pragma unroll 1
    for (int t0 = 0; t0 + 3 < T; t0 += 4) {
        // batched loads for the 4-step group
        const int4 ch4 = *(const int4*)(sptr + t0);
        const int  ch[4] = {ch4.x, ch4.y, ch4.z, ch4.w};
        // uniform L2 prefetch ~16 steps ahead (feats fits in 192MB L2)
        {
            int tp = t0 + 16; if (tp > T - 1) tp = T - 1;
            __builtin_prefetch(fbase + (size_t)tp * 8, 0, 0);
        }
#pragma unroll
        for (int u = 0; u < 4; ++u) {
            const int t = t0 + u;
            const float2 flo = *(const float2*)(fbase + (size_t)t * 8);
            const float2 fhi = *(const float2*)(fbase + (size_t)t * 8 + 4);
            crf_step(a1, a2, b1, b2, ch[u], flo, fhi, m, hi);
        }
        // renormalize once per 4 steps (safe: |gain| within ~1e14 .. 1e-40)
        float part = b1.x + b1.y + b2.x + b2.y;
        float tot  = part + __shfl(part, lane ^ 16, 32);
        tot = fmaxf(tot, 1e-30f);
        float inv = __builtin_amdgcn_rcpf(tot);
        b1.x *= inv; b1.y *= inv; b2.x *= inv; b2.y *= inv;
        sumlog2 += __builtin_amdgcn_logf(tot);   // log2; convert once at end
    }
    // tail (T not a multiple of 4)
    for (int t = T & ~3; t < T; ++t) {
        const int ch = sptr[t];
        const float2 flo = *(const float2*)(fbase + (size_t)t * 8);
        const float2 fhi = *(const float2*)(fbase + (size_t)t * 8 + 4);
        crf_step(a1, a2, b1, b2, ch, flo, fhi, m, hi);
        float part = b1.x + b1.y + b2.x + b2.y;
        float tot  = part + __shfl(part, lane ^ 16, 32);
        tot = fmaxf(tot, 1e-30f);
        float inv = __builtin_amdgcn_rcpf(tot);
        b1.x *= inv; b1.y *= inv; b2.x *= inv; b2.y *= inv;
        sumlog2 += __builtin_amdgcn_logf(tot);
    }

    // fwd = ln2*sumlog2 + T*mmax + logsumexp_j(log P[j] + M_unch[STOP, j])
    float st0 = wsA[129], st1 = wsA[130], st2 = wsA[131], st3 = wsA[132];
    float st4 = wsA[133], st5 = wsA[134], st6 = wsA[135], st7 = wsA[136];
    float stopm = fmaxf(fmaxf(fmaxf(st0, st1), fmaxf(st2, st3)),
                        fmaxf(fmaxf(st4, st5), fmaxf(st6, st7)));
    float e0 = __expf((hi ? st2 : st0) - stopm);
    float e1 = __expf((hi ? st3 : st1) - stopm);
    float e4 = __expf((hi ? st6 : st4) - stopm);
    float e5 = __expf((hi ? st7 : st5) - stopm);
    float part = b1.x * e0 + b1.y * e1 + b2.x * e4 + b2.y * e5;
    float tot  = part + __shfl(part, lane ^ 16, 32);
    tot = fmaxf(tot, 1e-30f);
    float fwd = 0.69314718056f * (sumlog2 + __builtin_amdgcn_logf(tot))
              + (float)T * mmax + stopm;

    if (!hi && valid) out[b] = fwd - gold[b];
}

// ---------------------------------------------------------------------------
extern "C" void kernel_launch(void* const* d_in, const int* in_sizes, int n_in,
                              void* d_out, int out_size, void* d_ws, size_t ws_size,
                              hipStream_t stream) {
    const float* feats = (const float*)d_in[0];
    const int*   spk   = (const int*)d_in[1];
    const int*   emos  = (const int*)d_in[2];
    const float* Ti    = (const float*)d_in[3];
    const float* Ta    = (const float*)d_in[4];
    const float* wui   = (const float*)d_in[5];
    const float* wua   = (const float*)d_in[6];
    const float* wci   = (const float*)d_in[7];
    const float* wca   = (const float*)d_in[8];
    float* out = (float*)d_out;
    float* ws  = (float*)d_ws;

    const int B = out_size;                 // 4096
    const int T = in_sizes[1] / B;          // 512

    crf_setup_kernel<<<1, 32, 0, stream>>>(Ti, Ta, wui, wua, wci, wca, ws);
    crf_gold_kernel<<<B, 256, 0, stream>>>(feats, spk, emos, Ti, Ta,
                                           wui, wua, wci, wca, ws + 256, T);
    const int tiles  = (B + 15) / 16;       // 16 batch rows per wave
    const int blocks = (tiles + 3) / 4;     // 4 waves (128 threads) per block
    crf_fwd_kernel<<<blocks, 128, 0, stream>>>(feats, spk, ws, ws + 256, out, B, T);
}